// GAT_21818433863880
// MI455X (gfx1250) — compile-verified
//
#include <hip/hip_runtime.h>

#define B_ 8
#define N_ 2048
#define IN_ 128
#define OUT_ 64
#define H_ 4
#define NEG_SLOPE_ 0.2f

typedef float v2f __attribute__((ext_vector_type(2)));
typedef float v8f __attribute__((ext_vector_type(8)));

__device__ __forceinline__ float lrelu(float v) {
    return v > 0.0f ? v : NEG_SLOPE_ * v;
}

// D = A(16x4 f32) * B(4x16 f32) + C(16x16 f32), wave32
__device__ __forceinline__ v8f wmma_f32(v2f a, v2f b, v8f c) {
    return __builtin_amdgcn_wmma_f32_16x16x4_f32(
        /*neg_a=*/false, a, /*neg_b=*/false, b,
        /*c_mod=*/(short)0, c, /*reuse_a=*/false, /*reuse_b=*/false);
}

// ---------------------------------------------------------------------------
// Kernel 1: h[b,h,n,o] = x[b,n,:] @ W[h,:,o] + bias[h,o]
//           fi[b,h,n]  = h . a1[h,:],  fj[b,h,n] = h . a2[h,:]
// Grid: B*(N/16) blocks, 128 threads (4 waves, wave w == head w), 16-row tile.
// ---------------------------------------------------------------------------
__global__ __launch_bounds__(128) void gat_h_fi_fj(
    const float* __restrict__ x, const float* __restrict__ W,
    const float* __restrict__ bias, const float* __restrict__ a,
    float* __restrict__ h_ws, float* __restrict__ fi, float* __restrict__ fj)
{
    const int blk  = blockIdx.x;
    const int b    = blk >> 7;            // / (N_/16)
    const int n0   = (blk & 127) << 4;
    const int head = threadIdx.x >> 5;    // wave index == head
    const int lane = threadIdx.x & 31;
    const int row  = lane & 15;
    const int half = lane >> 4;
    const int k0   = half << 1;           // K offset of this half-wave
    const int bh   = b * H_ + head;

    const float* xrow = x + (size_t)(b * N_ + n0 + row) * IN_;
    const float* Wh   = W + (size_t)head * IN_ * OUT_;

    v8f acc[4] = {};                      // 4 col-groups of 16 -> 64 outputs

    #pragma unroll 4
    for (int c = 0; c < IN_ / 4; ++c) {   // K in chunks of 4
        const int k = 4 * c + k0;
        v2f av; av.x = xrow[k]; av.y = xrow[k + 1];
        #pragma unroll
        for (int g = 0; g < 4; ++g) {
            const int col = g * 16 + row;
            v2f bv;
            bv.x = Wh[(size_t)k * OUT_ + col];
            bv.y = Wh[(size_t)(k + 1) * OUT_ + col];
            acc[g] = wmma_f32(av, bv, acc[g]);
        }
    }

    // + bias
    #pragma unroll
    for (int g = 0; g < 4; ++g) {
        const float bb = bias[head * OUT_ + g * 16 + row];
        #pragma unroll
        for (int r = 0; r < 8; ++r) acc[g][r] += bb;
    }

    // store h tile
    float* hb = h_ws + ((size_t)bh * N_ + n0) * OUT_;
    #pragma unroll
    for (int r = 0; r < 8; ++r) {
        const int orow = r + 8 * half;
        #pragma unroll
        for (int g = 0; g < 4; ++g)
            hb[(size_t)orow * OUT_ + g * 16 + row] = acc[g][r];
    }

    // fi/fj from accumulator layout: lane holds col (g*16 + row) of rows r+8*half
    float a1g[4], a2g[4];
    #pragma unroll
    for (int g = 0; g < 4; ++g) {
        a1g[g] = a[head * 2 * OUT_ + g * 16 + row];
        a2g[g] = a[head * 2 * OUT_ + OUT_ + g * 16 + row];
    }
    #pragma unroll
    for (int r = 0; r < 8; ++r) {
        float s1 = 0.0f, s2 = 0.0f;
        #pragma unroll
        for (int g = 0; g < 4; ++g) {
            s1 += acc[g][r] * a1g[g];
            s2 += acc[g][r] * a2g[g];
        }
        // reduce across the 16 lanes of this half (xor <= 8 stays in-half)
        #pragma unroll
        for (int off = 8; off; off >>= 1) {
            s1 += __shfl_xor(s1, off, 32);
            s2 += __shfl_xor(s2, off, 32);
        }
        if ((lane & 15) == 0) {
            const int orow = r + 8 * half;
            fi[(size_t)bh * N_ + n0 + orow] = s1;
            fj[(size_t)bh * N_ + n0 + orow] = s2;
        }
    }
}

// ---------------------------------------------------------------------------
// Kernel 2: fjmax[bh] = max_m fj[bh,m]   (lrelu monotonic => exact softmax max)
// ---------------------------------------------------------------------------
__global__ __launch_bounds__(256) void gat_fjmax(
    const float* __restrict__ fj, float* __restrict__ fjmax)
{
    __shared__ float red[256];
    const int bh = blockIdx.x;
    const int t  = threadIdx.x;
    float m = -3.4e38f;
    for (int i = t; i < N_; i += 256)
        m = fmaxf(m, fj[(size_t)bh * N_ + i]);
    red[t] = m;
    __syncthreads();
    for (int s = 128; s; s >>= 1) {
        if (t < s) red[t] = fmaxf(red[t], red[t + s]);
        __syncthreads();
    }
    if (t == 0) fjmax[bh] = red[0];
}

// ---------------------------------------------------------------------------
// Kernel 3: single-pass attention (rowmax known up-front), head-mean via LDS.
// Grid: B*(N/16) blocks, 128 threads (wave w == head w). Per wave: 16 query
// rows, loop over 128 m-tiles; P built in WMMA A-layout, O += P @ H_tile.
// ---------------------------------------------------------------------------
__global__ __launch_bounds__(128) void gat_attn(
    const float* __restrict__ h_ws, const float* __restrict__ fi,
    const float* __restrict__ fj, const float* __restrict__ fjmax,
    float* __restrict__ out)
{
    __shared__ float lds[H_][16][OUT_];   // 16 KB

    const int blk  = blockIdx.x;
    const int b    = blk >> 7;
    const int n0   = (blk & 127) << 4;
    const int head = threadIdx.x >> 5;
    const int lane = threadIdx.x & 31;
    const int row  = lane & 15;
    const int half = lane >> 4;
    const int k0   = half << 1;
    const int bh   = b * H_ + head;

    const float  fi_r   = fi[(size_t)bh * N_ + n0 + row];
    const float  rowmax = lrelu(fi_r + fjmax[bh]);
    const float* fjb    = fj + (size_t)bh * N_;
    const float* hb     = h_ws + (size_t)bh * N_ * OUT_;

    v8f   O[4] = {};
    float rowsum = 0.0f;

    for (int mt = 0; mt < N_ / 16; ++mt) {
        const int m0 = mt * 16;

        // prefetch next 16x64 h tile (global_prefetch_b8); 128B per lane
        if (mt + 1 < N_ / 16)
            __builtin_prefetch(&hb[(size_t)(m0 + 16) * OUT_ + lane * 32], 0, 1);

        // P tile in A-layout: lane -> row L%16, K = m0 + 4c + k0 + {0,1}
        v2f p[4];
        #pragma unroll
        for (int c = 0; c < 4; ++c) {
            const int k = m0 + 4 * c + k0;
            const float e0 = __expf(lrelu(fi_r + fjb[k])     - rowmax);
            const float e1 = __expf(lrelu(fi_r + fjb[k + 1]) - rowmax);
            p[c].x = e0; p[c].y = e1;
            rowsum += e0 + e1;
        }

        // O += P @ H_tile  (H in B-layout: lane -> col, rows = K split)
        #pragma unroll
        for (int g = 0; g < 4; ++g) {
            const int col = g * 16 + row;
            #pragma unroll
            for (int c = 0; c < 4; ++c) {
                const int k = m0 + 4 * c + k0;
                v2f bv;
                bv.x = hb[(size_t)k * OUT_ + col];
                bv.y = hb[(size_t)(k + 1) * OUT_ + col];
                O[g] = wmma_f32(p[c], bv, O[g]);
            }
        }
    }

    // combine the two half-wave partial row sums (each half covered K%4 halves)
    rowsum += __shfl_xor(rowsum, 16, 32);
    const float inv = 1.0f / rowsum;

    #pragma unroll
    for (int r = 0; r < 8; ++r) {
        const int   orow = r + 8 * half;
        const float sc   = __shfl(inv, orow, 32);  // lane orow holds row orow's inv
        #pragma unroll
        for (int g = 0; g < 4; ++g)
            lds[head][orow][g * 16 + row] = O[g][r] * sc;
    }
    __syncthreads();

    // mean over heads; 16*64 outputs / 128 threads = 8 each (deterministic)
    const int t = threadIdx.x;
    #pragma unroll
    for (int i = 0; i < 8; ++i) {
        const int idx  = t * 8 + i;
        const int orow = idx >> 6;
        const int col  = idx & 63;
        const float s = 0.25f * (lds[0][orow][col] + lds[1][orow][col] +
                                 lds[2][orow][col] + lds[3][orow][col]);
        out[(size_t)(b * N_ + n0 + orow) * OUT_ + col] = s;
    }
}

// ---------------------------------------------------------------------------
extern "C" void kernel_launch(void* const* d_in, const int* in_sizes, int n_in,
                              void* d_out, int out_size, void* d_ws, size_t ws_size,
                              hipStream_t stream) {
    (void)in_sizes; (void)n_in; (void)out_size; (void)ws_size;
    const float* x    = (const float*)d_in[0];   // (8,2048,128)
    const float* W    = (const float*)d_in[1];   // (4,128,64)
    const float* bias = (const float*)d_in[2];   // (4,64)
    const float* a    = (const float*)d_in[3];   // (4,128)
    float*       out  = (float*)d_out;           // (8,2048,64)

    float* wsf  = (float*)d_ws;
    float* h_ws = wsf;                                       // 16 MB (L2-resident)
    float* fi   = h_ws + (size_t)B_ * H_ * N_ * OUT_;        // 256 KB
    float* fj   = fi   + (size_t)B_ * H_ * N_;               // 256 KB
    float* fjm  = fj   + (size_t)B_ * H_ * N_;               // 128 B

    const int nblocks = B_ * (N_ / 16);  // 1024
    gat_h_fi_fj<<<nblocks, 128, 0, stream>>>(x, W, bias, a, h_ws, fi, fj);
    gat_fjmax  <<<B_ * H_, 256, 0, stream>>>(fj, fjm);
    gat_attn   <<<nblocks, 128, 0, stream>>>(h_ws, fi, fj, fjm, out);
}